// ClusterOverlap_83262236000463
// MI455X (gfx1250) — compile-verified
//
#include <hip/hip_runtime.h>
#include <math.h>

// ---------------------------------------------------------------------------
// ClusterOverlap on MI455X (gfx1250, wave32)
//   d2(i,j) = max(sq_i + sq_j - 2*Gram(i,j), 0),  Gram = E @ E^T
//   thr2_i  = 26th smallest d2 of row i (k=25 neighbours + self; sqrt is
//             monotone so selection/threshold/mask on d2 == reference on d)
//   counts  = (d2 < thr2) @ onehot(argmax(categorical)) -> entropy per row
// Gram tiles via V_WMMA_F32_16X16X4_F32; B-fragments register double-buffered
// so global loads overlap the WMMA accumulation chain.
// ---------------------------------------------------------------------------

typedef float v2f __attribute__((ext_vector_type(2)));
typedef float v8f __attribute__((ext_vector_type(8)));

#define B_ROWS 16384
#define DIMS   64
#define NCLUST 25
#define LISTN  26          // k+1 smallest (self included)
#define TILE   16
#define NKSTEP (DIMS / 4)  // 16 WMMA k-steps of K=4
#define TPAD   20          // LDS tile row stride (floats), 16B-aligned float4

// ---------------------------------------------------------------------------
// Kernel 0: zero the populated-cluster flags (ws is poisoned by harness)
// ---------------------------------------------------------------------------
__global__ void init_kernel(int* __restrict__ populated) {
    int t = threadIdx.x;
    if (t < NCLUST) populated[t] = 0;
}

// ---------------------------------------------------------------------------
// Kernel 1: per-row prep — copy encodings to out, row norms, argmax cluster,
// confident-cluster OR-reduction.
// ---------------------------------------------------------------------------
__global__ void prep_kernel(const float* __restrict__ E,
                            const float* __restrict__ cat,
                            float* __restrict__ out_enc,
                            float* __restrict__ sq,
                            int* __restrict__ clusters,
                            int* __restrict__ populated) {
    int j = blockIdx.x * blockDim.x + threadIdx.x;
    if (j >= B_ROWS) return;

    const float4* row  = (const float4*)(E + (size_t)j * DIMS);
    float4*       orow = (float4*)(out_enc + (size_t)j * DIMS);
    float s = 0.0f;
#pragma unroll
    for (int t = 0; t < DIMS / 4; ++t) {
        float4 v = row[t];
        orow[t]  = v;
        s += v.x * v.x + v.y * v.y + v.z * v.z + v.w * v.w;
    }
    sq[j] = s;

    const float* c = cat + (size_t)j * NCLUST;
    float mx = c[0];
    int   idx = 0;
#pragma unroll
    for (int q = 1; q < NCLUST; ++q) {
        float v = c[q];
        if (v > mx) { mx = v; idx = q; }   // first-occurrence argmax
    }
    clusters[j] = idx;
    if (mx >= 0.25f) atomicOr(&populated[idx], 1);
}

// ---------------------------------------------------------------------------
// Kernel 2: n_components = number of populated clusters
// ---------------------------------------------------------------------------
__global__ void ncomp_kernel(const int* __restrict__ populated,
                             float* __restrict__ out_ncomp) {
    if (threadIdx.x == 0) {
        int s = 0;
#pragma unroll
        for (int c = 0; c < NCLUST; ++c) s += (populated[c] != 0) ? 1 : 0;
        *out_ncomp = (float)s;
    }
}

// ---------------------------------------------------------------------------
// Pass 1: per-row squared threshold = 26th smallest squared distance.
// One wave per 16-row strip; A-fragments resident in registers for the whole
// 1024-tile column sweep; B-fragments double-buffered in registers.
// ---------------------------------------------------------------------------
__global__ void __launch_bounds__(32)
pass1_kernel(const float* __restrict__ E,
             const float* __restrict__ sq,
             float* __restrict__ thr2) {
    const int lane = threadIdx.x;
    const int i0   = blockIdx.x * TILE;
    const int m    = lane & 15;         // A row / B column index
    const int hi   = lane >> 4;         // selects K pair (and M+8 in C layout)
    const int half = hi << 1;           // K offset 0 or 2 within a k-step

    // A fragments: 16 rows x 64 K, resident for the whole sweep (32 VGPRs).
    const float* Abase = E + (size_t)(i0 + m) * DIMS + half;
    v2f a[NKSTEP];
#pragma unroll
    for (int kk = 0; kk < NKSTEP; ++kk) a[kk] = *(const v2f*)(Abase + kk * 4);

    float sqA[8];
#pragma unroll
    for (int v = 0; v < 8; ++v) sqA[v] = sq[i0 + v + 8 * hi];

    // Streaming 26-smallest list (squared distances), registers only.
    float lst[LISTN];
#pragma unroll
    for (int q = 0; q < LISTN; ++q) lst[q] = 3.4e38f;

    __shared__ __align__(16) float tile[TILE][TPAD];
    __shared__ float lists[32][LISTN];

    const int r  = lane >> 1;           // row this lane scans (selection)
    const int c0 = (lane & 1) * 8;      // its 8-column half of the tile

    const float* Bcol = E + (size_t)m * DIMS + half;  // + j0*DIMS per tile

    auto process = [&](int j0, const v2f* bf) {
        v8f acc = {};
#pragma unroll
        for (int kk = 0; kk < NKSTEP; ++kk)
            acc = __builtin_amdgcn_wmma_f32_16x16x4_f32(
                false, a[kk], false, bf[kk], (short)0, acc, false, false);

        float sqB = sq[j0 + m];
        // C layout: acc[v] = Gram[M = v + 8*hi][N = m]
#pragma unroll
        for (int v = 0; v < 8; ++v)
            tile[v + 8 * hi][m] = fmaxf(sqA[v] + sqB - 2.0f * acc[v], 0.0f);
        __syncthreads();

        // Row-major remap: two b128 LDS reads, insert 8 candidates.
        float4 ca = *(const float4*)&tile[r][c0];
        float4 cb = *(const float4*)&tile[r][c0 + 4];
        float cand[8] = {ca.x, ca.y, ca.z, ca.w, cb.x, cb.y, cb.z, cb.w};
#pragma unroll
        for (int t = 0; t < 8; ++t) {
            float x = cand[t];
            if (x < lst[LISTN - 1]) {   // wave skips bubble if no lane hits
                float cur = x;
#pragma unroll
                for (int q = 0; q < LISTN; ++q) {
                    float lo = fminf(cur, lst[q]);
                    cur      = fmaxf(cur, lst[q]);
                    lst[q]   = lo;
                }
            }
        }
        __syncthreads();
    };

    // Double-buffered column sweep: loads for tile t+1 overlap WMMAs of t.
    v2f b0[NKSTEP], b1[NKSTEP];
#pragma unroll
    for (int kk = 0; kk < NKSTEP; ++kk) b0[kk] = *(const v2f*)(Bcol + kk * 4);

    for (int j0 = 0; j0 < B_ROWS; j0 += 2 * TILE) {
        {
            const float* Bn = Bcol + (size_t)(j0 + TILE) * DIMS;
#pragma unroll
            for (int kk = 0; kk < NKSTEP; ++kk) b1[kk] = *(const v2f*)(Bn + kk * 4);
        }
        process(j0, b0);
        {
            int jp = (j0 + 2 * TILE < B_ROWS) ? (j0 + 2 * TILE) : 0;  // clamp
            const float* Bn = Bcol + (size_t)jp * DIMS;
#pragma unroll
            for (int kk = 0; kk < NKSTEP; ++kk) b0[kk] = *(const v2f*)(Bn + kk * 4);
        }
        process(j0 + TILE, b1);
    }

    // Merge the two half-row lists; 26th smallest of the union = thr2.
#pragma unroll
    for (int q = 0; q < LISTN; ++q) lists[lane][q] = lst[q];
    __syncthreads();
    if (lane < TILE) {
        int ia = 0, ib = 0;
        float x = 0.0f;
        for (int s = 0; s < LISTN; ++s) {
            float va = lists[2 * lane][ia];
            float vb = lists[2 * lane + 1][ib];
            if (va <= vb) { x = va; ++ia; } else { x = vb; ++ib; }
        }
        thr2[i0 + lane] = x;
    }
}

// ---------------------------------------------------------------------------
// Pass 2: recompute Gram tiles, mask d2 < thr2, LDS-atomic cluster histogram,
// then per-row Shannon entropy. Consumes the WMMA C layout directly.
// ---------------------------------------------------------------------------
__global__ void __launch_bounds__(32)
pass2_kernel(const float* __restrict__ E,
             const float* __restrict__ sq,
             const float* __restrict__ thr2,
             const int* __restrict__ clusters,
             float* __restrict__ entropy) {
    const int lane = threadIdx.x;
    const int i0   = blockIdx.x * TILE;
    const int m    = lane & 15;
    const int hi   = lane >> 4;
    const int half = hi << 1;

    const float* Abase = E + (size_t)(i0 + m) * DIMS + half;
    v2f a[NKSTEP];
#pragma unroll
    for (int kk = 0; kk < NKSTEP; ++kk) a[kk] = *(const v2f*)(Abase + kk * 4);

    float sqA[8], thrA[8];
#pragma unroll
    for (int v = 0; v < 8; ++v) {
        sqA[v]  = sq[i0 + v + 8 * hi];
        thrA[v] = thr2[i0 + v + 8 * hi];
    }

    __shared__ int cnt[TILE][NCLUST];
    for (int t = lane; t < TILE * NCLUST; t += 32) ((int*)cnt)[t] = 0;
    __syncthreads();

    const float* Bcol = E + (size_t)m * DIMS + half;

    auto process = [&](int j0, const v2f* bf) {
        v8f acc = {};
#pragma unroll
        for (int kk = 0; kk < NKSTEP; ++kk)
            acc = __builtin_amdgcn_wmma_f32_16x16x4_f32(
                false, a[kk], false, bf[kk], (short)0, acc, false, false);
        float sqB = sq[j0 + m];
        int   cl  = clusters[j0 + m];     // cluster of column N = m
#pragma unroll
        for (int v = 0; v < 8; ++v) {
            float d2 = fmaxf(sqA[v] + sqB - 2.0f * acc[v], 0.0f);
            if (d2 < thrA[v]) atomicAdd(&cnt[v + 8 * hi][cl], 1);  // ds_add
        }
    };

    v2f b0[NKSTEP], b1[NKSTEP];
#pragma unroll
    for (int kk = 0; kk < NKSTEP; ++kk) b0[kk] = *(const v2f*)(Bcol + kk * 4);

    for (int j0 = 0; j0 < B_ROWS; j0 += 2 * TILE) {
        {
            const float* Bn = Bcol + (size_t)(j0 + TILE) * DIMS;
#pragma unroll
            for (int kk = 0; kk < NKSTEP; ++kk) b1[kk] = *(const v2f*)(Bn + kk * 4);
        }
        process(j0, b0);
        {
            int jp = (j0 + 2 * TILE < B_ROWS) ? (j0 + 2 * TILE) : 0;  // clamp
            const float* Bn = Bcol + (size_t)jp * DIMS;
#pragma unroll
            for (int kk = 0; kk < NKSTEP; ++kk) b0[kk] = *(const v2f*)(Bn + kk * 4);
        }
        process(j0 + TILE, b1);
    }
    __syncthreads();

    if (lane < TILE) {
        int n = 0;
#pragma unroll
        for (int c = 0; c < NCLUST; ++c) n += cnt[lane][c];
        float inv = 1.0f / (float)n;      // n >= 1 (self always counted)
        float e = 0.0f;
#pragma unroll
        for (int c = 0; c < NCLUST; ++c) {
            float b = (float)cnt[lane][c] * inv;
            e -= b * logf(b + 1e-5f);     // zero bins contribute exactly 0
        }
        entropy[i0 + lane] = e;
    }
}

// ---------------------------------------------------------------------------
extern "C" void kernel_launch(void* const* d_in, const int* in_sizes, int n_in,
                              void* d_out, int out_size, void* d_ws, size_t ws_size,
                              hipStream_t stream) {
    const float* E   = (const float*)d_in[0];   // [16384, 64]  fp32
    const float* cat = (const float*)d_in[1];   // [16384, 25]  fp32

    float* out       = (float*)d_out;
    float* out_enc   = out;                                   // B*D floats
    float* out_ent   = out + (size_t)B_ROWS * DIMS;           // B floats
    float* out_ncomp = out + (size_t)B_ROWS * DIMS + B_ROWS;  // 1 float

    // Workspace layout (~192 KB): sq | thr2 | clusters | populated
    float* sq        = (float*)d_ws;
    float* thr2      = sq + B_ROWS;
    int*   clusters  = (int*)(thr2 + B_ROWS);
    int*   populated = clusters + B_ROWS;

    init_kernel<<<1, 32, 0, stream>>>(populated);
    prep_kernel<<<B_ROWS / 128, 128, 0, stream>>>(E, cat, out_enc, sq,
                                                  clusters, populated);
    ncomp_kernel<<<1, 32, 0, stream>>>(populated, out_ncomp);
    pass1_kernel<<<B_ROWS / TILE, 32, 0, stream>>>(E, sq, thr2);
    pass2_kernel<<<B_ROWS / TILE, 32, 0, stream>>>(E, sq, thr2, clusters, out_ent);
}